// GraphSAGEBaseline_10187662426182
// MI455X (gfx1250) — compile-verified
//
#include <hip/hip_runtime.h>
#include <math.h>

typedef float v2f __attribute__((ext_vector_type(2)));
typedef float v8f __attribute__((ext_vector_type(8)));

#define D        128
#define HD       64
#define NLAYERS  3
#define BN_EPS   1e-5f

// ---------------------------------------------------------------- utilities
__global__ void zero_f(float* __restrict__ p, int n) {
    int i = blockIdx.x * blockDim.x + threadIdx.x;
    if (i < n) p[i] = 0.0f;
}

__global__ void count_edges(const int* __restrict__ dst, int E, float* __restrict__ cnt) {
    int e = blockIdx.x * blockDim.x + threadIdx.x;
    if (e < E) unsafeAtomicAdd(&cnt[dst[e]], 1.0f);
}

__global__ void inv_counts(float* __restrict__ cnt, int n) {
    int i = blockIdx.x * blockDim.x + threadIdx.x;
    if (i < n) cnt[i] = 1.0f / fmaxf(cnt[i], 1.0f);
}

// ------------------------------------------------- edge scatter (mean aggr)
// one thread per (edge, 4-float chunk): float4 gather from src row,
// 4 hardware fp32 atomics into dst row. 32 chunks cover D=128.
__global__ void scatter_feats(const float* __restrict__ h,
                              const int* __restrict__ src,
                              const int* __restrict__ dst,
                              int E, float* __restrict__ agg) {
    int idx = blockIdx.x * blockDim.x + threadIdx.x;
    int total = E * (D / 4);
    if (idx >= total) return;
    int e = idx >> 5;
    int c = (idx & 31) * 4;
    int s = src[e];
    int d = dst[e];
    const float4 v = *reinterpret_cast<const float4*>(h + (size_t)s * D + c);
    float* b = agg + (size_t)d * D + c;
    unsafeAtomicAdd(b + 0, v.x);
    unsafeAtomicAdd(b + 1, v.y);
    unsafeAtomicAdd(b + 2, v.z);
    unsafeAtomicAdd(b + 3, v.w);
}

// ------------------------------------------- fused SAGE layer GEMM (WMMA f32)
// y[N,128] = (agg*invc) @ Wl + h @ Wr + bl ; also accumulates per-column
// sum / sum-of-squares for BatchNorm via 2 atomics per lane.
// grid.x = N/16 row tiles; 256 threads = 8 waves, wave w owns col tile w.
__global__ __launch_bounds__(256)
void sage_gemm_wmma(const float* __restrict__ agg, const float* __restrict__ invc,
                    const float* __restrict__ h,
                    const float* __restrict__ Wl, const float* __restrict__ Wr,
                    const float* __restrict__ bl,
                    float* __restrict__ y,
                    float* __restrict__ colsum, float* __restrict__ colsum2) {
    const int wave = threadIdx.x >> 5;
    const int lane = threadIdx.x & 31;
    const int half = lane >> 4;       // selects K pair for A/B
    const int l15  = lane & 15;
    const int rowBase = blockIdx.x * 16;
    const int col = wave * 16 + l15;  // this lane's B/C column
    const int arow = rowBase + l15;   // this lane's A row
    const float ic = invc[arow];
    const float* aggRow = agg + (size_t)arow * D;
    const float* hRow   = h   + (size_t)arow * D;

    v8f c = {};
#pragma unroll 4
    for (int kb = 0; kb < D; kb += 4) {
        const int ka = kb + half * 2;
        const float2 a2 = *reinterpret_cast<const float2*>(aggRow + ka);
        const float2 h2 = *reinterpret_cast<const float2*>(hRow + ka);
        v2f aL, aR, bL, bR;
        aL.x = a2.x * ic;                 aL.y = a2.y * ic;
        aR.x = h2.x;                      aR.y = h2.y;
        bL.x = Wl[(size_t)ka * D + col];  bL.y = Wl[(size_t)(ka + 1) * D + col];
        bR.x = Wr[(size_t)ka * D + col];  bR.y = Wr[(size_t)(ka + 1) * D + col];
        c = __builtin_amdgcn_wmma_f32_16x16x4_f32(false, aL, false, bL, (short)0, c, false, false);
        c = __builtin_amdgcn_wmma_f32_16x16x4_f32(false, aR, false, bR, (short)0, c, false, false);
    }

    const float bias = bl[col];
    float s = 0.0f, s2 = 0.0f;
#pragma unroll
    for (int v = 0; v < 8; ++v) {
        float val = c[v] + bias;
        s  += val;
        s2 += val * val;
        const int r = rowBase + half * 8 + v;   // C layout: VGPR v -> row v (+8 for upper half)
        y[(size_t)r * D + col] = val;
    }
    unsafeAtomicAdd(&colsum[col],  s);
    unsafeAtomicAdd(&colsum2[col], s2);
}

// ----------------------------------------------------- BN params + BN/ReLU
__global__ void bn_params(const float* __restrict__ colsum, const float* __restrict__ colsum2,
                          const float* __restrict__ gamma, const float* __restrict__ beta,
                          float invN, float* __restrict__ scale, float* __restrict__ shift) {
    int c = threadIdx.x;
    float mu  = colsum[c] * invN;
    float var = colsum2[c] * invN - mu * mu;   // biased variance
    float sc  = gamma[c] / sqrtf(var + BN_EPS);
    scale[c] = sc;
    shift[c] = beta[c] - mu * sc;
}

__global__ void bn_relu(float* __restrict__ y, const float* __restrict__ scale,
                        const float* __restrict__ shift, int n) {
    int i = blockIdx.x * blockDim.x + threadIdx.x;
    if (i < n) {
        int col = i & (D - 1);
        y[i] = fmaxf(fmaf(y[i], scale[col], shift[col]), 0.0f);
    }
}

// -------------------------------------------------- MLP head: z = relu(h@W1+b1)
// grid.x = N/16; 128 threads = 4 waves covering 64 output cols.
__global__ __launch_bounds__(128)
void head_gemm_wmma(const float* __restrict__ h, const float* __restrict__ W1,
                    const float* __restrict__ b1, float* __restrict__ z) {
    const int wave = threadIdx.x >> 5;
    const int lane = threadIdx.x & 31;
    const int half = lane >> 4;
    const int l15  = lane & 15;
    const int rowBase = blockIdx.x * 16;
    const int col  = wave * 16 + l15;          // 0..63
    const int arow = rowBase + l15;
    const float* hRow = h + (size_t)arow * D;

    v8f c = {};
#pragma unroll 4
    for (int kb = 0; kb < D; kb += 4) {
        const int ka = kb + half * 2;
        const float2 a2 = *reinterpret_cast<const float2*>(hRow + ka);
        v2f a, b;
        a.x = a2.x;                        a.y = a2.y;
        b.x = W1[(size_t)ka * HD + col];   b.y = W1[(size_t)(ka + 1) * HD + col];
        c = __builtin_amdgcn_wmma_f32_16x16x4_f32(false, a, false, b, (short)0, c, false, false);
    }
    const float bias = b1[col];
#pragma unroll
    for (int v = 0; v < 8; ++v) {
        const int r = rowBase + half * 8 + v;
        z[(size_t)r * HD + col] = fmaxf(c[v] + bias, 0.0f);
    }
}

// ---------------------------------------- final 64x2 dot + softmax, per row
__global__ void head_out(const float* __restrict__ z, const float* __restrict__ W2,
                         const float* __restrict__ b2, float* __restrict__ out, int n) {
    int row = blockIdx.x * blockDim.x + threadIdx.x;
    if (row >= n) return;
    float a0 = b2[0], a1 = b2[1];
    const float* zr = z + (size_t)row * HD;
#pragma unroll
    for (int k = 0; k < HD; ++k) {
        float v = zr[k];
        a0 = fmaf(v, W2[k * 2 + 0], a0);
        a1 = fmaf(v, W2[k * 2 + 1], a1);
    }
    out[(size_t)row * 2 + 0] = a0;
    out[(size_t)row * 2 + 1] = a1;
    float m  = fmaxf(a0, a1);
    float e0 = expf(a0 - m);
    float e1 = expf(a1 - m);
    float inv = 1.0f / (e0 + e1);
    out[(size_t)n * 2 + (size_t)row * 2 + 0] = e0 * inv;
    out[(size_t)n * 2 + (size_t)row * 2 + 1] = e1 * inv;
}

// ---------------------------------------------------------------- launcher
extern "C" void kernel_launch(void* const* d_in, const int* in_sizes, int n_in,
                              void* d_out, int out_size, void* d_ws, size_t ws_size,
                              hipStream_t stream) {
    const float* x     = (const float*)d_in[0];
    const int*   ei    = (const int*)  d_in[1];
    const float* Wl    = (const float*)d_in[2];
    const float* Wr    = (const float*)d_in[3];
    const float* bl    = (const float*)d_in[4];
    const float* gamma = (const float*)d_in[5];
    const float* beta  = (const float*)d_in[6];
    const float* W1    = (const float*)d_in[7];
    const float* b1    = (const float*)d_in[8];
    const float* W2    = (const float*)d_in[9];
    const float* b2    = (const float*)d_in[10];

    const int N = in_sizes[0] / D;
    const int E = in_sizes[1] / 2;
    const int* src = ei;
    const int* dst = ei + E;

    const size_t nd = (size_t)N * D;
    float* ws      = (float*)d_ws;
    float* hA      = ws;  ws += nd;
    float* hB      = ws;  ws += nd;
    float* agg     = ws;  ws += nd;   // reused as z for the head
    float* cnt     = ws;  ws += N;
    float* colsum  = ws;  ws += D;    // colsum2 must immediately follow
    float* colsum2 = ws;  ws += D;
    float* bnscale = ws;  ws += D;
    float* bnshift = ws;  ws += D;

    const int ndi = (int)nd;

    // in-degree counts (dst is invariant across layers)
    zero_f     <<<(N + 255) / 256, 256, 0, stream>>>(cnt, N);
    count_edges<<<(E + 255) / 256, 256, 0, stream>>>(dst, E, cnt);
    inv_counts <<<(N + 255) / 256, 256, 0, stream>>>(cnt, N);

    const float* hcur = x;
    float* ybuf = hA;
    for (int l = 0; l < NLAYERS; ++l) {
        zero_f<<<(ndi + 255) / 256, 256, 0, stream>>>(agg, ndi);
        zero_f<<<1, 256, 0, stream>>>(colsum, 2 * D);
        const int scThreads = E * (D / 4);
        scatter_feats<<<(scThreads + 255) / 256, 256, 0, stream>>>(hcur, src, dst, E, agg);
        sage_gemm_wmma<<<N / 16, 256, 0, stream>>>(agg, cnt, hcur,
                Wl + (size_t)l * D * D, Wr + (size_t)l * D * D, bl + (size_t)l * D,
                ybuf, colsum, colsum2);
        bn_params<<<1, D, 0, stream>>>(colsum, colsum2, gamma + (size_t)l * D,
                                       beta + (size_t)l * D, 1.0f / (float)N,
                                       bnscale, bnshift);
        bn_relu<<<(ndi + 255) / 256, 256, 0, stream>>>(ybuf, bnscale, bnshift, ndi);
        hcur = ybuf;
        ybuf = (ybuf == hA) ? hB : hA;
    }

    float* z = agg;  // reuse aggregation buffer
    head_gemm_wmma<<<N / 16, 128, 0, stream>>>(hcur, W1, b1, z);
    head_out<<<(N + 255) / 256, 256, 0, stream>>>(z, W2, b2, (float*)d_out, N);
}